// Rel_ds_31370441130197
// MI455X (gfx1250) — compile-verified
//
#include <hip/hip_runtime.h>
#include <math.h>

typedef float v2f __attribute__((ext_vector_type(2)));
typedef float v8f __attribute__((ext_vector_type(8)));

#define WMMA4(A,B,C) __builtin_amdgcn_wmma_f32_16x16x4_f32( \
    false, (A), false, (B), (short)0, (C), false, false)

// TIME_SCALE = 30000/500*0.3 = 18.0 ; ln(10000) = 9.210340371976184
__global__ __launch_bounds__(256) void relmink_wmma_kernel(
    const float* __restrict__ pos, const float* __restrict__ dom_time,
    const float* __restrict__ W, const float* __restrict__ bias,
    float* __restrict__ rel_out, float* __restrict__ emb_out,
    int numTiles)
{
  const int lane = threadIdx.x & 31;
  const int n    = lane & 15;   // row (pair-in-tile) for A; output column for B/D
  const int hi   = lane >> 4;   // half-wave selector: K offset +2 for A and B
  const int gwave  = blockIdx.x * (blockDim.x >> 5) + (threadIdx.x >> 5);
  const int nWaves = gridDim.x * (blockDim.x >> 5);

  // ---- per-wave constants: W^T operand, identity operand, freqs, bias ----
  // K pattern per lane: K(s,v) = 4*s + v + 2*hi  (covers K=0..31 over s=0..7)
  v2f wB0[8], wB1[8], iB0[8], iB1[8];
  float fq[8][2];
  #pragma unroll
  for (int s = 0; s < 8; ++s) {
    #pragma unroll
    for (int v = 0; v < 2; ++v) {
      const int K = 4*s + v + 2*hi;
      const int j = K & 15;
      const float f = __expf((float)j * (-9.210340371976184f / 16.0f));
      const float w0 = W[(n      ) * 32 + K];   // W[e][K], e = n      (cols 0..15)
      const float w1 = W[(16 + n ) * 32 + K];   // W[e][K], e = 16 + n (cols 16..31)
      const float i0 = (K == n      ) ? 1.0f : 0.0f;
      const float i1 = (K == 16 + n ) ? 1.0f : 0.0f;
      fq[s][v] = f;
      if (v == 0) { wB0[s].x = w0; wB1[s].x = w1; iB0[s].x = i0; iB1[s].x = i1; }
      else        { wB0[s].y = w0; wB1[s].y = w1; iB0[s].y = i0; iB1[s].y = i1; }
    }
  }
  const float bias0 = bias[n];
  const float bias1 = bias[16 + n];

  // ---- grid-stride over 16-pair tiles (wave-uniform loop) ----
  for (int t = gwave; t < numTiles; t += nWaves) {
    const int base  = t << 4;
    const int p     = base + n;           // this lane's pair (both halves compute it)
    const int bl    = p >> 10;            // flat (b*L + l)
    const int bidx  = p >> 20;            // batch index (L = 1024)
    const int mflat = (bidx << 10) | (p & 1023);  // flat (b*L + m)

    const float dx = pos[bl*3 + 0] - pos[mflat*3 + 0];
    const float dy = pos[bl*3 + 1] - pos[mflat*3 + 1];
    const float dz = pos[bl*3 + 2] - pos[mflat*3 + 2];
    const float dt = (dom_time[bl] - dom_time[mflat]) * 18.0f;
    const float ds2 = dx*dx + dy*dy + dz*dz - dt*dt;
    const float d   = copysignf(sqrtf(fabsf(ds2)), ds2);
    const float x   = 1024.0f * fminf(4.0f, fmaxf(-4.0f, d));

    // A operand: this lane's half of the 16x32 emb tile (dims K(s,v))
    // K < 16 (s<4) -> sin ; K >= 16 (s>=4) -> cos of same freq index (K & 15)
    v2f A[8];
    #pragma unroll
    for (int s = 0; s < 8; ++s) {
      if (s < 4) { A[s].x = __sinf(x * fq[s][0]); A[s].y = __sinf(x * fq[s][1]); }
      else       { A[s].x = __cosf(x * fq[s][0]); A[s].y = __cosf(x * fq[s][1]); }
    }

    v8f cW0 = {bias0,bias0,bias0,bias0,bias0,bias0,bias0,bias0};
    v8f cW1 = {bias1,bias1,bias1,bias1,bias1,bias1,bias1,bias1};
    v8f cE0 = {};
    v8f cE1 = {};
    #pragma unroll
    for (int s = 0; s < 8; ++s) {
      cW0 = WMMA4(A[s], wB0[s], cW0);   // rel_attn cols  0..15
      cW1 = WMMA4(A[s], wB1[s], cW1);   // rel_attn cols 16..31
      cE0 = WMMA4(A[s], iB0[s], cE0);   // emb transpose, dims  0..15
      cE1 = WMMA4(A[s], iB1[s], cE1);   // emb transpose, dims 16..31
    }

    // D layout: acc[j] of lane -> row (base + j + 8*hi), col (blk*16 + n)
    // -> 64B-contiguous stores per half-wave
    #pragma unroll
    for (int j = 0; j < 8; ++j) {
      const size_t row = (size_t)(base + j + 8*hi);
      rel_out[row*32 + n]      = cW0[j];
      rel_out[row*32 + 16 + n] = cW1[j];
      emb_out[row*32 + n]      = cE0[j];
      emb_out[row*32 + 16 + n] = cE1[j];
    }
  }
}

extern "C" void kernel_launch(void* const* d_in, const int* in_sizes, int n_in,
                              void* d_out, int out_size, void* d_ws, size_t ws_size,
                              hipStream_t stream) {
  (void)n_in; (void)out_size; (void)d_ws; (void)ws_size;
  const float* pos      = (const float*)d_in[0];
  const float* dom_time = (const float*)d_in[1];
  const float* W        = (const float*)d_in[2];
  const float* b        = (const float*)d_in[3];

  const int L  = 1024;                 // fixed by the reference
  const int BL = in_sizes[1];          // B*L
  const int B  = BL / L;
  const long long totalPairs = (long long)B * L * L;
  const int numTiles = (int)(totalPairs / 16);

  float* rel = (float*)d_out;
  float* emb = rel + (size_t)totalPairs * 32;

  hipLaunchKernelGGL(relmink_wmma_kernel, dim3(1024), dim3(256), 0, stream,
                     pos, dom_time, W, b, rel, emb, numTiles);
}